// QuantumLayer_88587995447972
// MI455X (gfx1250) — compile-verified
//
#include <hip/hip_runtime.h>
#include <hip/hip_bf16.h>

#define MM      16      // modes
#define NP      8       // photons
#define NS      12870   // C(16,8)
#define NSUB    256     // 2^NP
#define NB      8       // batch
#define NOUT    128

#define KBLK_PER_B 51   // ceil(12870/256)
#define NCHUNK  32      // split-K blocks for the WMMA GEMM
#define KCHUNK  404     // multiple of 4; 32*404 = 12928 >= 12870

typedef __attribute__((ext_vector_type(2))) float v2f;
typedef __attribute__((ext_vector_type(8))) float v8f;

__device__ __forceinline__ float2 cmul(float2 a, float2 b) {
    return make_float2(a.x * b.x - a.y * b.y, a.x * b.y + a.y * b.x);
}
__device__ __forceinline__ float2 cadd(float2 a, float2 b) {
    return make_float2(a.x + b.x, a.y + b.y);
}

// ---------------------------------------------------------------------------
// Kernel 1: build V[b][p][s] = sum_{n in subset s} Uin[b][p][n]
//   A = M1 @ diag(e^{i th1}) @ M0   (columns 0..7 only)
//   B = M3 @ diag(e^{i th2}) @ M2
//   Uin[b][p][n] = sum_m B[p][m] * e^{i x[b][m]} * A[m][n]
// ---------------------------------------------------------------------------
__global__ void build_V(const float* __restrict__ x,
                        const float* __restrict__ th1,
                        const float* __restrict__ th2,
                        const float* __restrict__ mr,
                        const float* __restrict__ mi,
                        float2* __restrict__ V)
{
    __shared__ float2 ph1[16];
    __shared__ float2 ph2[16];
    __shared__ float2 phx[NB * 16];
    __shared__ float2 sA[16 * 8];      // A[m][n], n < 8
    __shared__ float2 sB[16 * 16];     // B[p][m]
    __shared__ float2 sU[NB * 16 * 8]; // Uin[b][p][n]

    const int t = threadIdx.x;

    if (t < 16) {
        float s, c; sincosf(th1[t], &s, &c); ph1[t] = make_float2(c, s);
    } else if (t < 32) {
        float s, c; sincosf(th2[t - 16], &s, &c); ph2[t - 16] = make_float2(c, s);
    }
    if (t >= 32 && t < 32 + NB * 16) {
        int i = t - 32;
        float s, c; sincosf(x[i], &s, &c); phx[i] = make_float2(c, s);
    }
    __syncthreads();

    { // B: 256 entries, one per thread
        int p = t >> 4, m = t & 15;
        float2 acc = make_float2(0.f, 0.f);
        #pragma unroll
        for (int tt = 0; tt < 16; ++tt) {
            float2 M3 = make_float2(mr[3 * 256 + p * 16 + tt], mi[3 * 256 + p * 16 + tt]);
            float2 M2 = make_float2(mr[2 * 256 + tt * 16 + m], mi[2 * 256 + tt * 16 + m]);
            acc = cadd(acc, cmul(cmul(M3, ph2[tt]), M2));
        }
        sB[p * 16 + m] = acc;
    }
    if (t < 128) { // A: 16x8 entries
        int m = t >> 3, n = t & 7;
        float2 acc = make_float2(0.f, 0.f);
        #pragma unroll
        for (int tt = 0; tt < 16; ++tt) {
            float2 M1 = make_float2(mr[1 * 256 + m * 16 + tt], mi[1 * 256 + m * 16 + tt]);
            float2 M0 = make_float2(mr[0 * 256 + tt * 16 + n], mi[0 * 256 + tt * 16 + n]);
            acc = cadd(acc, cmul(cmul(M1, ph1[tt]), M0));
        }
        sA[m * 8 + n] = acc;
    }
    __syncthreads();

    // Uin: 8*16*8 = 1024 entries, 4 per thread
    for (int e = t; e < NB * 16 * 8; e += 256) {
        int b = e >> 7, p = (e >> 3) & 15, n = e & 7;
        float2 acc = make_float2(0.f, 0.f);
        #pragma unroll
        for (int m = 0; m < 16; ++m)
            acc = cadd(acc, cmul(cmul(sB[p * 16 + m], phx[b * 16 + m]), sA[m * 8 + n]));
        sU[e] = acc;
    }
    __syncthreads();

    // V: 8*16*256 entries, 128 per thread
    for (int e = t; e < NB * 16 * NSUB; e += 256) {
        int b = e >> 12, p = (e >> 8) & 15, s = e & 255;
        const float2* u = &sU[(b * 16 + p) * 8];
        float2 acc = make_float2(0.f, 0.f);
        #pragma unroll
        for (int n = 0; n < 8; ++n)
            if ((s >> n) & 1) acc = cadd(acc, u[n]);
        V[e] = acc;
    }
}

// ---------------------------------------------------------------------------
// Kernel 2: Ryser permanents.
//   probs[b][k] = | sum_s sign(s) * prod_i V[b][rows[k][i]][s] |^2
// Block = 256 combos; V[b] cached in LDS, rows padded to 257 float2
// (stride 514 dwords -> bank shift of 2 per row, spreads 64-bank conflicts).
// ---------------------------------------------------------------------------
__global__ void permanents(const float2* __restrict__ V,
                           const int* __restrict__ rows_idx,
                           float* __restrict__ probs)
{
    __shared__ float2 lds[16 * 257];
    const int b = blockIdx.x / KBLK_PER_B;
    const int kbase = (blockIdx.x % KBLK_PER_B) * 256;
    const int t = threadIdx.x;

    for (int i = t; i < 16 * 256; i += 256) {
        int p = i >> 8, s = i & 255;
        lds[p * 257 + s] = V[(b * 16 + p) * 256 + s];
    }
    __syncthreads();

    const int k = kbase + t;
    if (k >= NS) return;

    int r[8];
    #pragma unroll
    for (int i = 0; i < 8; ++i) r[i] = rows_idx[k * 8 + i] * 257;

    float sre = 0.f, sim = 0.f;
    for (int s = 0; s < NSUB; ++s) {
        float2 p = lds[r[0] + s];
        #pragma unroll
        for (int i = 1; i < 8; ++i) p = cmul(p, lds[r[i] + s]);
        float sg = (__popc(s) & 1) ? -1.f : 1.f;   // * (-1)^NP == * 1
        sre = fmaf(sg, p.x, sre);
        sim = fmaf(sg, p.y, sim);
    }
    probs[b * NS + k] = sre * sre + sim * sim;
}

// ---------------------------------------------------------------------------
// Kernel 3: deterministic per-batch sum -> inv_sum[b] = 1 / sum_k probs[b][k]
// ---------------------------------------------------------------------------
__global__ void rowsum(const float* __restrict__ probs, float* __restrict__ inv_sum)
{
    __shared__ float red[256];
    const int b = blockIdx.x, t = threadIdx.x;
    float s = 0.f;
    for (int k = t; k < NS; k += 256) s += probs[b * NS + k];
    red[t] = s;
    __syncthreads();
    for (int w = 128; w > 0; w >>= 1) {
        if (t < w) red[t] += red[t + w];
        __syncthreads();
    }
    if (t == 0) inv_sum[b] = 1.0f / red[0];
}

// ---------------------------------------------------------------------------
// Kernel 4: split-K WMMA partial GEMM.
//   partial[c][m][n] = sum_{k in chunk c} probs[m][k] * weight[k][n]
// Grid = NCHUNK blocks x 256 threads (8 waves = 8 N-tiles of 16).
// f32 WMMA 16x16x4; M=8 padded to 16 (pad rows alias row 0 -- harmless,
// since A row m only contributes to C row m and rows 8..15 are discarded).
// Main loop is branch/guard-free; 2-element K tail handled once via
// index-clamped unconditional loads (v_cndmask, no exec masking).
// ---------------------------------------------------------------------------
__global__ void gemm_wmma_partial(const float* __restrict__ probs,
                                  const float* __restrict__ weight,
                                  float* __restrict__ partial)
{
    const int wave = threadIdx.x >> 5;          // 0..7 -> N tile
    const int lane = threadIdx.x & 31;
    const int n0   = wave * 16;
    const int row  = lane & 15;                 // A row (M) and B column (N)
    const int kh   = (lane >> 4) << 1;          // 0 | 2: K sub-offset per half-wave

    const int kstart = blockIdx.x * KCHUNK;
    const int kend   = min(kstart + KCHUNK, NS);

    const int arow = (row < NB) ? row : 0;      // pad rows alias row 0
    const float* __restrict__ prow = probs + (size_t)arow * NS;
    const float* __restrict__ wcol = weight + n0 + row;

    const float* ap = prow + kstart + kh;                     // A walker (b64 loads)
    const float* wp = wcol + (size_t)(kstart + kh) * NOUT;    // B walker (2x b32)

    v8f c = {};
    int k0 = kstart;
    #pragma unroll 4
    for (; k0 + 4 <= kend; k0 += 4) {
        v2f a = *(const v2f*)ap;                // ka even -> 8B aligned
        v2f b;
        b.x = wp[0];
        b.y = wp[NOUT];
        __builtin_prefetch(wp + (size_t)256 * NOUT, 0, 1);  // global_prefetch_b8
        c = __builtin_amdgcn_wmma_f32_16x16x4_f32(
                false, a, false, b, (short)0, c, false, false);
        ap += 4;
        wp += (size_t)4 * NOUT;
    }
    if (k0 < kend) {                            // block-uniform tail (<=3 elems)
        const int ka = k0 + kh;
        const float m0 = (ka     < kend) ? 1.0f : 0.0f;
        const float m1 = (ka + 1 < kend) ? 1.0f : 0.0f;
        const int   i0 = (ka     < kend) ? ka     : k0;      // clamped, in-bounds
        const int   i1 = (ka + 1 < kend) ? ka + 1 : k0;
        v2f a, b;
        a.x = prow[i0] * m0;
        a.y = prow[i1] * m1;
        b.x = wcol[(size_t)i0 * NOUT];
        b.y = wcol[(size_t)i1 * NOUT];
        c = __builtin_amdgcn_wmma_f32_16x16x4_f32(
                false, a, false, b, (short)0, c, false, false);
    }

    // C layout: VGPR j -> M=j (lanes 0-15) / M=j+8 (lanes 16-31).
    if (lane < 16) {
        float* dst = partial + ((size_t)blockIdx.x * NB) * NOUT + n0 + row;
        #pragma unroll
        for (int j = 0; j < NB; ++j)
            dst[j * NOUT] = c[j];
    }
}

// ---------------------------------------------------------------------------
// Kernel 5: deterministic split-K reduction + normalize + bias.
//   out[m][n] = inv_sum[m] * sum_c partial[c][m][n] + bias[n]
// ---------------------------------------------------------------------------
__global__ void reduce_out(const float* __restrict__ partial,
                           const float* __restrict__ inv_sum,
                           const float* __restrict__ bias,
                           float* __restrict__ out)
{
    const int t = threadIdx.x;            // 1024 threads = 8*128 outputs
    const int m = t >> 7, n = t & 127;
    float s = 0.f;
    for (int cidx = 0; cidx < NCHUNK; ++cidx)        // fixed order: deterministic
        s += partial[((size_t)cidx * NB + m) * NOUT + n];
    out[m * NOUT + n] = s * inv_sum[m] + bias[n];
}

// ---------------------------------------------------------------------------
extern "C" void kernel_launch(void* const* d_in, const int* in_sizes, int n_in,
                              void* d_out, int out_size, void* d_ws, size_t ws_size,
                              hipStream_t stream) {
    const float* x        = (const float*)d_in[0];  // 8*16
    const float* theta1   = (const float*)d_in[1];  // 16
    const float* theta2   = (const float*)d_in[2];  // 16
    const float* m_real   = (const float*)d_in[3];  // 4*16*16
    const float* m_imag   = (const float*)d_in[4];  // 4*16*16
    const float* weight   = (const float*)d_in[5];  // 12870*128
    const float* bias     = (const float*)d_in[6];  // 128
    const int*   rows_idx = (const int*)d_in[7];    // 12870*8

    float*  ws       = (float*)d_ws;
    float2* V        = (float2*)ws;                          // 65536 floats
    float*  probs    = ws + (size_t)NB * 16 * NSUB * 2;      // 102960 floats
    float*  inv_sum  = probs + (size_t)NB * NS;              // 8 floats
    float*  partial  = inv_sum + NB;                         // 32*8*128 = 32768 floats

    build_V<<<1, 256, 0, stream>>>(x, theta1, theta2, m_real, m_imag, V);
    permanents<<<NB * KBLK_PER_B, 256, 0, stream>>>(V, rows_idx, probs);
    rowsum<<<NB, 256, 0, stream>>>(probs, inv_sum);
    gemm_wmma_partial<<<NCHUNK, 256, 0, stream>>>(probs, weight, partial);
    reduce_out<<<1, 1024, 0, stream>>>(partial, inv_sum, bias, (float*)d_out);
}